// ComplexFractalRegularizationLoss_67027259621646
// MI455X (gfx1250) — compile-verified
//
#include <hip/hip_runtime.h>

typedef float v2f __attribute__((ext_vector_type(2)));
typedef float v8f __attribute__((ext_vector_type(8)));

// Wave32 full reduction using the CDNA5 matrix pipe.
// A (16x4 f32): lanes 0-15 hold row M=lane {K0 in VGPR0, K1 in VGPR1},
// lanes 16-31 hold {K2, K3} for rows 0-15. With a.y==0 and B==ones,
// D[m][n] = s[m] + s[m+16] for every n. D layout: VGPR r holds M=r
// (lanes 0-15) / M=8+r (lanes 16-31). Summing the 8 D VGPRs per lane and
// xor-16 shuffle gives the total in every lane.
__device__ __forceinline__ float wave_sum_wmma(float s) {
    v2f a; a.x = s;   a.y = 0.0f;
    v2f b; b.x = 1.0f; b.y = 1.0f;
    v8f c = {0.f, 0.f, 0.f, 0.f, 0.f, 0.f, 0.f, 0.f};
    c = __builtin_amdgcn_wmma_f32_16x16x4_f32(false, a, false, b,
                                              (short)0, c, false, false);
    float t = ((c[0] + c[1]) + (c[2] + c[3])) + ((c[4] + c[5]) + (c[6] + c[7]));
    t += __shfl_xor(t, 16, 32);
    return t;
}

// Streaming pass: per-block partial abs-sum and partial nonzero-count
// (count restricted to indices < limit, matching the box-count remainder drop).
__global__ void frac_reduce_kernel(const float* __restrict__ src, int n, int limit,
                                   float* __restrict__ out_sum,
                                   float* __restrict__ out_cnt) {
    const int nvec   = n >> 2;
    const int tail   = n & 3;
    const int stride = gridDim.x * blockDim.x;
    const int gtid   = blockIdx.x * blockDim.x + threadIdx.x;

    float s = 0.0f, cnt = 0.0f;
    const float4* v = (const float4*)src;
    for (int i = gtid; i < nvec; i += stride) {
        float4 x = v[i];                       // global_load_b128
        float a0 = fabsf(x.x), a1 = fabsf(x.y);
        float a2 = fabsf(x.z), a3 = fabsf(x.w);
        s += (a0 + a1) + (a2 + a3);
        int base = i << 2;
        cnt += ((a0 > 0.0f) && (base     < limit)) ? 1.0f : 0.0f;
        cnt += ((a1 > 0.0f) && (base + 1 < limit)) ? 1.0f : 0.0f;
        cnt += ((a2 > 0.0f) && (base + 2 < limit)) ? 1.0f : 0.0f;
        cnt += ((a3 > 0.0f) && (base + 3 < limit)) ? 1.0f : 0.0f;
    }
    if (gtid < tail) {                         // reconverges before wmma
        int idx = (nvec << 2) + gtid;
        float av = fabsf(src[idx]);
        s += av;
        cnt += ((av > 0.0f) && (idx < limit)) ? 1.0f : 0.0f;
    }

    float wsum = wave_sum_wmma(s);             // EXEC all ones here
    float wcnt = wave_sum_wmma(cnt);

    __shared__ float lsum[8], lcnt[8];
    int lane = threadIdx.x & 31, wid = threadIdx.x >> 5;
    if (lane == 0) { lsum[wid] = wsum; lcnt[wid] = wcnt; }
    __syncthreads();
    if (threadIdx.x == 0) {
        int nw = blockDim.x >> 5;
        float bs = 0.0f, bc = 0.0f;
        for (int w = 0; w < nw; ++w) { bs += lsum[w]; bc += lcnt[w]; }
        out_sum[blockIdx.x] = bs;
        out_cnt[blockIdx.x] = bc;
    }
}

// Finish pass: deterministically reduce the 4 partial arrays, apply the
// scalar formula with lambda/epoch/total_epochs read from device memory.
__global__ void frac_final_kernel(const float* __restrict__ ws, int nblk,
                                  const float* __restrict__ lam_p,
                                  const int* __restrict__ epoch_p,
                                  const int* __restrict__ total_p,
                                  float nb1_inv, float nb2_inv,
                                  float* __restrict__ out) {
    const float* s1 = ws;
    const float* c1 = ws + nblk;
    const float* s2 = ws + 2 * nblk;
    const float* c2 = ws + 3 * nblk;

    float ps1 = 0.f, pc1 = 0.f, ps2 = 0.f, pc2 = 0.f;
    for (int i = threadIdx.x; i < nblk; i += blockDim.x) {
        ps1 += s1[i]; pc1 += c1[i]; ps2 += s2[i]; pc2 += c2[i];
    }

    float S1 = wave_sum_wmma(ps1);
    float C1 = wave_sum_wmma(pc1);
    float S2 = wave_sum_wmma(ps2);
    float C2 = wave_sum_wmma(pc2);

    __shared__ float l[4][8];
    int lane = threadIdx.x & 31, wid = threadIdx.x >> 5;
    if (lane == 0) { l[0][wid] = S1; l[1][wid] = C1; l[2][wid] = S2; l[3][wid] = C2; }
    __syncthreads();

    if (threadIdx.x == 0) {
        float t0 = 0.f, t1 = 0.f, t2 = 0.f, t3 = 0.f;
        int nw = blockDim.x >> 5;
        for (int w = 0; w < nw; ++w) {
            t0 += l[0][w]; t1 += l[1][w]; t2 += l[2][w]; t3 += l[3][w];
        }
        float lam      = *lam_p;
        float progress = 1.0f - (float)(*epoch_p) / (float)(*total_p);
        // fractal_loss = ((0 + S1^2)*lam + S2^2)*lam   (faithful accumulation order)
        float fractal = ((t0 * t0) * lam + t2 * t2) * lam;
        float m1 = t1 * nb1_inv;
        float m2 = t3 * nb2_inv;
        float mr = m1 * m1 + m2 * m2;          // stop_gradient irrelevant for fwd value
        *out = (0.1f * progress) * fractal + (0.01f * progress) * mr;
    }
}

extern "C" void kernel_launch(void* const* d_in, const int* in_sizes, int n_in,
                              void* d_out, int out_size, void* d_ws, size_t ws_size,
                              hipStream_t stream) {
    const float* W1    = (const float*)d_in[0];
    const float* W2    = (const float*)d_in[1];
    const float* lam   = (const float*)d_in[2];
    const int*   epoch = (const int*)d_in[3];
    const int*   total = (const int*)d_in[4];
    float*       out   = (float*)d_out;

    const int N1 = in_sizes[0];
    const int N2 = in_sizes[1];

    // 2048 blocks x 256 threads: ~16 float4 loads per thread per matrix;
    // enough blocks to saturate 23.3 TB/s HBM, workspace = 4*nblk floats.
    int nblk = 2048;
    while ((size_t)4 * (size_t)nblk * sizeof(float) > ws_size && nblk > 1) nblk >>= 1;

    float* wsf = (float*)d_ws;
    const int limit1 = (N1 / 3) * 3;
    const int limit2 = (N2 / 3) * 3;

    frac_reduce_kernel<<<nblk, 256, 0, stream>>>(W1, N1, limit1, wsf,            wsf + nblk);
    frac_reduce_kernel<<<nblk, 256, 0, stream>>>(W2, N2, limit2, wsf + 2 * nblk, wsf + 3 * nblk);

    const float nb1_inv = 1.0f / (float)(N1 / 3);
    const float nb2_inv = 1.0f / (float)(N2 / 3);
    frac_final_kernel<<<1, 256, 0, stream>>>(wsf, nblk, lam, epoch, total,
                                             nb1_inv, nb2_inv, out);
}